// ShortRangeModel_52158082842761
// MI455X (gfx1250) — compile-verified
//
#include <hip/hip_runtime.h>
#include <math.h>

// ---------------------------------------------------------------------------
// ShortRangeModel for MI455X (gfx1250, wave32).
// Phase 1: masked pairwise RBF features (transcendental-light: exp/cos only
//          for the ~0.8% of pairs inside the cutoff).
// Phase 2: MLP 16->64->64->1 via V_WMMA_F32_16X16X4_F32 (fp32 WMMA).
// Phase 3: deterministic reduction of per-block energy sums.
// ---------------------------------------------------------------------------

typedef __attribute__((ext_vector_type(2))) float v2f;
typedef __attribute__((ext_vector_type(8))) float v8f;

#define N_ATOMS   4096
#define NRBF      16
#define NHID      64
#define CUTOFF_F  5.0f
#define RMIN_F    0.5f
#define PI_F      3.14159265358979323846f

// ----------------------------- Phase 1 -------------------------------------
// One block (256 threads = 8 waves) per atom i; threads stride over j.
__global__ __launch_bounds__(256)
void feats_kernel(const float* __restrict__ pos,
                  const float* __restrict__ centers,
                  float* __restrict__ feats)
{
    __shared__ float sC[NRBF];
    __shared__ float red[256 * NRBF];   // 16 KB reduction scratch

    const int i   = blockIdx.x;
    const int tid = threadIdx.x;

    if (tid < NRBF) sC[tid] = centers[tid];
    __syncthreads();

    const float xi = pos[3 * i + 0];
    const float yi = pos[3 * i + 1];
    const float zi = pos[3 * i + 2];

    const float cut2    = CUTOFF_F * CUTOFF_F;
    const float eta     = 0.5f * (CUTOFF_F - RMIN_F) / (float)NRBF;  // 0.140625
    const float inv2e2  = 1.0f / (2.0f * eta * eta);
    const float invCut  = 1.0f / CUTOFF_F;

    float acc[NRBF];
#pragma unroll
    for (int k = 0; k < NRBF; ++k) acc[k] = 0.0f;

    for (int j = tid; j < N_ATOMS; j += 256) {
        const float dx = xi - pos[3 * j + 0];
        const float dy = yi - pos[3 * j + 1];
        const float dz = zi - pos[3 * j + 2];
        const float sq = dx * dx + dy * dy + dz * dz;
        if (sq > 0.0f && sq < cut2) {
            const float dist   = sqrtf(sq);
            const float x      = dist * invCut;
            const float smooth = 0.5f * (1.0f + __cosf(PI_F * x));
#pragma unroll
            for (int k = 0; k < NRBF; ++k) {
                const float d = dist - sC[k];
                acc[k] += smooth * __expf(-d * d * inv2e2);
            }
        }
    }

#pragma unroll
    for (int k = 0; k < NRBF; ++k) red[tid * NRBF + k] = acc[k];
    __syncthreads();

    for (int off = 128; off > 0; off >>= 1) {
        if (tid < off) {
#pragma unroll
            for (int k = 0; k < NRBF; ++k)
                red[tid * NRBF + k] += red[(tid + off) * NRBF + k];
        }
        __syncthreads();
    }

    if (tid < NRBF) feats[i * NRBF + tid] = red[tid];
}

// ----------------------------- Phase 2 -------------------------------------
// 256 threads = 8 waves per block; each wave owns a 16-atom row tile.
// Layouts per CDNA5 ISA (f32 16x16x4):
//   A (16x4):  lane L<16 -> row M=L, VGPR0=K0, VGPR1=K1; lanes 16-31 -> K2,K3
//   B (4x16):  VGPR0: lanes 0-15 = row K0, lanes 16-31 = row K2 (cols = lane&15)
//   C/D (16x16): VGPR v: lanes 0-15 -> M=v, lanes 16-31 -> M=v+8, N = lane&15
#define LDS_STRIDE 68   // 64 + 4 pad: column reads across 16 lanes hit distinct banks

__device__ __forceinline__ float silu_f(float x)
{
    return x / (1.0f + __expf(-x));
}

__global__ __launch_bounds__(256)
void mlp_kernel(const float* __restrict__ feats,
                const float* __restrict__ W1, const float* __restrict__ b1,
                const float* __restrict__ W2, const float* __restrict__ b2,
                const float* __restrict__ W3, const float* __restrict__ b3,
                float* __restrict__ blockSums)
{
    __shared__ float sh[8][16][LDS_STRIDE];  // per-wave 16x64 activation tile
    __shared__ float en[128];                // per-row energies for this block

    const int tid  = threadIdx.x;
    const int wave = tid >> 5;
    const int lane = tid & 31;
    const int m    = lane & 15;      // column (N) / row (M) selector
    const int half = lane >> 4;      // 0: K0,K1 / rows 0-7 ; 1: K2,K3 / rows 8-15
    const int kk   = half * 2;
    const int row0 = (blockIdx.x * 8 + wave) * 16;   // first atom of this tile

    // ---- layer 1: feats(16x16) @ W1(16x64) -> 4 tiles of 16x16, K in 4-chunks
    v8f acc[4] = {};
#pragma unroll
    for (int k0 = 0; k0 < NRBF; k0 += 4) {
        v2f a;
        a.x = feats[(row0 + m) * NRBF + k0 + kk + 0];
        a.y = feats[(row0 + m) * NRBF + k0 + kk + 1];
#pragma unroll
        for (int t = 0; t < 4; ++t) {
            v2f b;
            b.x = W1[(k0 + kk + 0) * NHID + t * 16 + m];
            b.y = W1[(k0 + kk + 1) * NHID + t * 16 + m];
            acc[t] = __builtin_amdgcn_wmma_f32_16x16x4_f32(
                false, a, false, b, (short)0, acc[t], false, false);
        }
    }

    // bias + SiLU, scatter D-layout into LDS as (row, col) tile
#pragma unroll
    for (int t = 0; t < 4; ++t) {
        const int   n    = t * 16 + m;
        const float bias = b1[n];
#pragma unroll
        for (int v = 0; v < 8; ++v) {
            const float h = silu_f(acc[t][v] + bias);
            sh[wave][v + 8 * half][n] = h;
        }
    }
    __syncthreads();

    // ---- layer 2: h1(16x64) @ W2(64x64), K=64 in 4-chunks
    v8f acc2[4] = {};
#pragma unroll
    for (int k0 = 0; k0 < NHID; k0 += 4) {
        v2f a;
        a.x = sh[wave][m][k0 + kk + 0];
        a.y = sh[wave][m][k0 + kk + 1];
#pragma unroll
        for (int t = 0; t < 4; ++t) {
            v2f b;
            b.x = W2[(k0 + kk + 0) * NHID + t * 16 + m];
            b.y = W2[(k0 + kk + 1) * NHID + t * 16 + m];
            acc2[t] = __builtin_amdgcn_wmma_f32_16x16x4_f32(
                false, a, false, b, (short)0, acc2[t], false, false);
        }
    }
    __syncthreads();   // everyone done reading h1 before overwrite

    // bias + SiLU, write h2 back to LDS
#pragma unroll
    for (int t = 0; t < 4; ++t) {
        const int   n    = t * 16 + m;
        const float bias = b2[n];
#pragma unroll
        for (int v = 0; v < 8; ++v) {
            const float h = silu_f(acc2[t][v] + bias);
            sh[wave][v + 8 * half][n] = h;
        }
    }
    __syncthreads();

    // ---- layer 3: per-atom energy = h2 . W3 + b3   (lanes 0-15, one row each)
    if (lane < 16) {
        float e = b3[0];
#pragma unroll 8
        for (int k = 0; k < NHID; ++k) e += sh[wave][lane][k] * W3[k];
        en[wave * 16 + lane] = e;
    }
    __syncthreads();

    // deterministic per-block sum (fixed order -> bitwise-stable replays)
    if (tid == 0) {
        float s = 0.0f;
        for (int r = 0; r < 128; ++r) s += en[r];
        blockSums[blockIdx.x] = s;
    }
}

// ----------------------------- Phase 3 -------------------------------------
__global__ void reduce_kernel(const float* __restrict__ blockSums,
                              float* __restrict__ out, int n)
{
    if (threadIdx.x == 0) {
        float s = 0.0f;
        for (int i = 0; i < n; ++i) s += blockSums[i];
        out[0] = s;
    }
}

// ---------------------------------------------------------------------------
extern "C" void kernel_launch(void* const* d_in, const int* in_sizes, int n_in,
                              void* d_out, int out_size, void* d_ws, size_t ws_size,
                              hipStream_t stream)
{
    (void)in_sizes; (void)n_in; (void)out_size; (void)ws_size;

    const float* positions = (const float*)d_in[0];   // (4096, 3)
    const float* centers   = (const float*)d_in[1];   // (16,)
    const float* W1        = (const float*)d_in[2];   // (16, 64)
    const float* b1        = (const float*)d_in[3];   // (64,)
    const float* W2        = (const float*)d_in[4];   // (64, 64)
    const float* b2        = (const float*)d_in[5];   // (64,)
    const float* W3        = (const float*)d_in[6];   // (64, 1)
    const float* b3        = (const float*)d_in[7];   // (1,)
    float*       out       = (float*)d_out;

    // workspace layout: feats (4096*16 f32) | blockSums (32 f32)
    float* feats     = (float*)d_ws;
    float* blockSums = feats + N_ATOMS * NRBF;

    feats_kernel<<<N_ATOMS, 256, 0, stream>>>(positions, centers, feats);

    const int mlpBlocks = N_ATOMS / 128;  // 8 waves * 16 rows per block
    mlp_kernel<<<mlpBlocks, 256, 0, stream>>>(feats, W1, b1, W2, b2, W3, b3,
                                              blockSums);

    reduce_kernel<<<1, 32, 0, stream>>>(blockSums, out, mlpBlocks);
}